// VolumetricRenderer_15006615733891
// MI455X (gfx1250) — compile-verified
//
#include <hip/hip_runtime.h>

// Volumetric renderer (NeRF compositing), MI455X / gfx1250.
//
// Memory-bound: ~3.84 KB read per ray, 252 MB total @ 23.3 TB/s => ~11 us floor.
// Strategy: one wave32 per ray; each lane owns 6 contiguous samples.
//  - sigma/t staged into LDS with CDNA5 async loads (ASYNCcnt), giving free
//    neighbor access for delta = t[n+1]-t[n].
//  - rgb prefetched (global_prefetch_b8) during the async copies, then read
//    directly during the weighted reduction.
//  - exclusive cumprod = 6 serial mults per lane + 5-step wave32 shfl scan.

#define N_SAMPLES 192
#define PER_LANE 6              // 192 / 32
#define WAVES_PER_BLOCK 8
#define THREADS (WAVES_PER_BLOCK * 32)
#define FAR_DELTA_F 1e10f
#define EPS_F 1e-10f

__global__ __launch_bounds__(THREADS) void volrend_wave32_kernel(
    const float* __restrict__ rgb,     // [B, N, 3]
    const float* __restrict__ sigma,   // [B, N]
    const float* __restrict__ t_vals,  // [B, N]
    float* __restrict__ out,           // [B*3] rgb_map then [B] depth_map
    int B)
{
    // Per-wave slice: t[192] then sigma[192]  (1536 B, 16B aligned)
    __shared__ __align__(16) float smem[WAVES_PER_BLOCK * 2 * N_SAMPLES];

    const int lane = threadIdx.x & 31;
    const int wv   = threadIdx.x >> 5;
    const int ray  = __builtin_amdgcn_readfirstlane(blockIdx.x * WAVES_PER_BLOCK + wv);
    if (ray >= B) return;

    float* wbase = &smem[wv * (2 * N_SAMPLES)];
    const unsigned lds_t = (unsigned)(unsigned long long)(void*)wbase;
    const unsigned lds_s = lds_t + N_SAMPLES * 4u;

    const unsigned long long tbase =
        (unsigned long long)(const void*)(t_vals + (size_t)ray * N_SAMPLES);
    const unsigned long long sbase =
        (unsigned long long)(const void*)(sigma + (size_t)ray * N_SAMPLES);

    // ---- CDNA5 async bulk copy: 192 floats each for t and sigma ----
    // b128: lanes cover floats [0,128) (16 B/lane); b64: floats [128,192) (8 B/lane).
    {
        unsigned g128 = (unsigned)lane * 16u;          // byte offset, 16B aligned
        unsigned g64  = 512u + (unsigned)lane * 8u;    // byte offset, 8B aligned
        unsigned lt128 = lds_t + g128, lt64 = lds_t + g64;
        unsigned ls128 = lds_s + g128, ls64 = lds_s + g64;
        asm volatile("global_load_async_to_lds_b128 %0, %1, %2"
                     :: "v"(lt128), "v"(g128), "s"(tbase) : "memory");
        asm volatile("global_load_async_to_lds_b64 %0, %1, %2"
                     :: "v"(lt64), "v"(g64), "s"(tbase) : "memory");
        asm volatile("global_load_async_to_lds_b128 %0, %1, %2"
                     :: "v"(ls128), "v"(g128), "s"(sbase) : "memory");
        asm volatile("global_load_async_to_lds_b64 %0, %1, %2"
                     :: "v"(ls64), "v"(g64), "s"(sbase) : "memory");
    }

    // Prefetch this lane's rgb chunk (18 floats, 72 B) while copies are in flight.
    const float* rgb_lane =
        rgb + (size_t)ray * (N_SAMPLES * 3) + (size_t)lane * (PER_LANE * 3);
    __builtin_prefetch(rgb_lane, 0, 3);

    // Wait for async copies into our wave's LDS slice.
    asm volatile("s_wait_asynccnt 0x0" ::: "memory");

    const int base = lane * PER_LANE;

    // Pull staged values; t[PER_LANE] is the first sample of the next lane
    // (in-bounds LDS read; unused when lane==31, which uses FAR_DELTA).
    float t[PER_LANE + 1];
    float sg[PER_LANE];
#pragma unroll
    for (int j = 0; j < PER_LANE; ++j) {
        t[j]  = wbase[base + j];
        sg[j] = wbase[N_SAMPLES + base + j];
    }
    t[PER_LANE] = wbase[base + PER_LANE];  // lane31 reads sigma[0]; ignored below

    // alpha_j = 1 - e, q_j = (1 - alpha + eps) = e + eps
    float a[PER_LANE], q[PER_LANE];
#pragma unroll
    for (int j = 0; j < PER_LANE; ++j) {
        float delta = (lane == 31 && j == PER_LANE - 1) ? FAR_DELTA_F
                                                        : (t[j + 1] - t[j]);
        float e = __expf(-sg[j] * delta);
        a[j] = 1.0f - e;
        q[j] = e + EPS_F;
    }

    // Lane-local product, then wave32 multiplicative Hillis-Steele scan.
    float p = ((q[0] * q[1]) * (q[2] * q[3])) * (q[4] * q[5]);
    float sc = p;
#pragma unroll
    for (int d = 1; d < 32; d <<= 1) {
        float u = __shfl_up(sc, d, 32);
        if (lane >= d) sc *= u;
    }
    float T = __shfl_up(sc, 1, 32);   // exclusive prefix product
    if (lane == 0) T = 1.0f;

    // Weighted accumulation: w_j = alpha_j * T; T *= q_j.
    float accR = 0.0f, accG = 0.0f, accB = 0.0f, accD = 0.0f;
#pragma unroll
    for (int j = 0; j < PER_LANE; ++j) {
        float w = a[j] * T;
        T *= q[j];
        accR += w * rgb_lane[j * 3 + 0];
        accG += w * rgb_lane[j * 3 + 1];
        accB += w * rgb_lane[j * 3 + 2];
        accD += w * t[j];
    }

    // Wave32 butterfly reduction.
#pragma unroll
    for (int off = 16; off > 0; off >>= 1) {
        accR += __shfl_down(accR, off, 32);
        accG += __shfl_down(accG, off, 32);
        accB += __shfl_down(accB, off, 32);
        accD += __shfl_down(accD, off, 32);
    }

    if (lane == 0) {
        out[(size_t)ray * 3 + 0] = accR;
        out[(size_t)ray * 3 + 1] = accG;
        out[(size_t)ray * 3 + 2] = accB;
        out[(size_t)B * 3 + ray] = accD;   // depth_map appended after rgb_map
    }
}

extern "C" void kernel_launch(void* const* d_in, const int* in_sizes, int n_in,
                              void* d_out, int out_size, void* d_ws, size_t ws_size,
                              hipStream_t stream) {
    const float* rgb    = (const float*)d_in[0];   // [B, N, 3]
    const float* sigma  = (const float*)d_in[1];   // [B, N]
    const float* t_vals = (const float*)d_in[2];   // [B, N]
    float* out = (float*)d_out;

    const int B = in_sizes[1] / N_SAMPLES;         // 65536 for the reference
    const int blocks = (B + WAVES_PER_BLOCK - 1) / WAVES_PER_BLOCK;

    hipLaunchKernelGGL(volrend_wave32_kernel, dim3(blocks), dim3(THREADS), 0, stream,
                       rgb, sigma, t_vals, out, B);
}